// LieHomographies_22067541967495
// MI455X (gfx1250) — compile-verified
//
#include <hip/hip_runtime.h>

typedef float v2f __attribute__((ext_vector_type(2)));
typedef float v8f __attribute__((ext_vector_type(8)));

#define WAVES_PER_BLOCK 4
#define BLOCK_T (WAVES_PER_BLOCK * 32)
#define ELEMS_PER_WAVE 64
#define ELEMS_PER_BLOCK (WAVES_PER_BLOCK * ELEMS_PER_WAVE)

static __device__ __forceinline__ v2f fma2(v2f a, v2f b, v2f c) {
    return __builtin_elementwise_fma(a, b, c);
}

// Two homographies per lane (packed in float2 -> v_pk_fma_f32), 64 per wave.
// WMMA f32 16x16x4 computes the (B,8)x(8,9) generator contraction; results are
// transposed from WMMA D-layout to per-lane layout through LDS.
__global__ __launch_bounds__(BLOCK_T) void lie_expm_wmma_kernel(
    const float* __restrict__ lie,    // (B, 8)
    const float* __restrict__ elems,  // (8, 3, 3)
    const float* __restrict__ wts,    // (8,)
    float* __restrict__ out,          // (B, 3, 3)
    int nB)
{
    // 12-float row stride: 9 used entries + padding to avoid bank conflicts.
    __shared__ float ldsG[WAVES_PER_BLOCK][ELEMS_PER_WAVE][12];

    const int lane = threadIdx.x & 31;
    const int wv   = threadIdx.x >> 5;
    const int blockBase = blockIdx.x * ELEMS_PER_BLOCK;
    if (blockBase >= nB) return;                 // whole-block uniform exit
    const int waveBase = blockBase + wv * ELEMS_PER_WAVE;

    const int  n  = lane & 15;                   // N column of WMMA tile
    const bool hi = lane >= 16;
    const int  k0 = hi ? 2 : 0;                  // K pair this lane supplies

    // ---- B operand: WE[k][n] = w[k] * elements[k][n/3][n%3], cols 9..15 = 0.
    // f32 B layout mirrors A: VGPR0 = K={0|2}, VGPR1 = K={1|3}; +4 for 2nd tile.
    v2f b0 = {0.f, 0.f}, b1 = {0.f, 0.f};
    if (n < 9) {
        b0.x = wts[k0]     * elems[(k0)     * 9 + n];
        b0.y = wts[k0 + 1] * elems[(k0 + 1) * 9 + n];
        b1.x = wts[k0 + 4] * elems[(k0 + 4) * 9 + n];
        b1.y = wts[k0 + 5] * elems[(k0 + 5) * 9 + n];
    }

    // ---- Four M-tiles of 16 elements -> G for 64 elements, scattered to LDS.
    #pragma unroll
    for (int t = 0; t < 4; ++t) {
        int row = waveBase + t * 16 + n;             // A-matrix row (batch idx)
        if (row >= nB) row = nB - 1;                 // clamp (tail safety)
        const float* pr = lie + row * 8 + k0;
        v2f a0 = { pr[0], pr[1] };                   // K = k0, k0+1
        v2f a1 = { pr[4], pr[5] };                   // K = k0+4, k0+5
        v8f acc = {0.f, 0.f, 0.f, 0.f, 0.f, 0.f, 0.f, 0.f};
        acc = __builtin_amdgcn_wmma_f32_16x16x4_f32(
                  false, a0, false, b0, (short)0, acc, false, false);
        acc = __builtin_amdgcn_wmma_f32_16x16x4_f32(
                  false, a1, false, b1, (short)0, acc, false, false);
        // D layout: VGPR v -> M = v (lanes 0-15) / v+8 (lanes 16-31), N = n.
        if (n < 9) {
            #pragma unroll
            for (int v = 0; v < 8; ++v)
                ldsG[wv][t * 16 + v + (hi ? 8 : 0)][n] = acc[v];
        }
    }
    __syncthreads();

    // ---- Per-lane expm of TWO matrices: element eA in .x, element eB in .y.
    // Every op below is a packed-f32 op on two independent problems.
    const int eA = waveBase + lane;
    const int eB = waveBase + 32 + lane;

    const float* ga = &ldsG[wv][lane][0];
    const float* gb = &ldsG[wv][32 + lane][0];
    const v2f sc2 = (v2f){1.0f / 256.0f, 1.0f / 256.0f};   // 2^-S, S = 8

    v2f As[9];
    #pragma unroll
    for (int d = 0; d < 9; ++d) As[d] = (v2f){ga[d], gb[d]} * sc2;

    v2f E[9];
    #pragma unroll
    for (int d = 0; d < 9; ++d) E[d] = (v2f){0.f, 0.f};
    E[0] = E[4] = E[8] = (v2f){1.f, 1.f};

    // Horner: E = I + (A @ E) / k, k = 12..1
    #pragma unroll
    for (int k = 12; k >= 1; --k) {
        v2f M[9];
        #pragma unroll
        for (int i = 0; i < 3; ++i) {
            #pragma unroll
            for (int j = 0; j < 3; ++j) {
                M[i * 3 + j] = fma2(As[i * 3 + 2], E[6 + j],
                               fma2(As[i * 3 + 1], E[3 + j],
                                    As[i * 3 + 0] * E[j]));
            }
        }
        const float invk  = 1.0f / (float)k;
        const v2f   invk2 = (v2f){invk, invk};
        #pragma unroll
        for (int d = 0; d < 9; ++d) {
            const v2f idd = (d == 0 || d == 4 || d == 8) ? (v2f){1.f, 1.f}
                                                         : (v2f){0.f, 0.f};
            E[d] = fma2(M[d], invk2, idd);
        }
    }

    // 8 squarings: E = E @ E
    #pragma unroll
    for (int s = 0; s < 8; ++s) {
        v2f M[9];
        #pragma unroll
        for (int i = 0; i < 3; ++i) {
            #pragma unroll
            for (int j = 0; j < 3; ++j) {
                M[i * 3 + j] = fma2(E[i * 3 + 2], E[6 + j],
                               fma2(E[i * 3 + 1], E[3 + j],
                                    E[i * 3 + 0] * E[j]));
            }
        }
        #pragma unroll
        for (int d = 0; d < 9; ++d) E[d] = M[d];
    }

    // ---- Homogeneous normalization + stores (lane extracts .x/.y for free).
    const v2f inv = (v2f){1.0f / E[8].x, 1.0f / E[8].y};
    v2f O[9];
    #pragma unroll
    for (int d = 0; d < 9; ++d) O[d] = E[d] * inv;

    if (eA < nB) {
        float* po = out + eA * 9;
        #pragma unroll
        for (int d = 0; d < 9; ++d) po[d] = O[d].x;
    }
    if (eB < nB) {
        float* po = out + eB * 9;
        #pragma unroll
        for (int d = 0; d < 9; ++d) po[d] = O[d].y;
    }
}

extern "C" void kernel_launch(void* const* d_in, const int* in_sizes, int n_in,
                              void* d_out, int out_size, void* d_ws, size_t ws_size,
                              hipStream_t stream) {
    const float* lie   = (const float*)d_in[0];   // (B, 8)
    const float* elems = (const float*)d_in[1];   // (8, 3, 3)
    const float* wts   = (const float*)d_in[2];   // (8,)
    float*       out   = (float*)d_out;           // (B, 3, 3)

    const int nB = in_sizes[0] / 8;
    const int grid = (nB + ELEMS_PER_BLOCK - 1) / ELEMS_PER_BLOCK;
    lie_expm_wmma_kernel<<<grid, BLOCK_T, 0, stream>>>(lie, elems, wts, out, nB);
}